// SubjectLayers_37194416783482
// MI455X (gfx1250) — compile-verified
//
#include <hip/hip_runtime.h>
#include <stdint.h>

// out[b,d,t] = sum_c x[b,c,t] * weights[subjects[b],c,d]
// B=64, C=256, D=256, T=4096, S=32. Per batch: (DxC)*(CxT) f32 GEMM via
// V_WMMA_F32_16X16X4_F32, with double-buffered async global->LDS panel
// staging (GLOBAL_LOAD_ASYNC_TO_LDS_B128, ASYNCcnt).

#define GB 64
#define GC 256
#define GD 256
#define GT 4096

#define KB 16              // channels per LDS stage
#define NSTAGE (GC / KB)   // 16 stages
#define WS 144             // W panel LDS row stride in dwords (128 + 16 pad)
#define XS 272             // X panel LDS row stride in dwords (256 + 16 pad)

typedef float v2f __attribute__((ext_vector_type(2)));
typedef float v8f __attribute__((ext_vector_type(8)));

// Async copy 16B global -> LDS, tracked by ASYNCcnt (no VGPR transit).
__device__ __forceinline__ void async_copy_b128(uint32_t lds_byte_off,
                                                const float* gaddr) {
    asm volatile("global_load_async_to_lds_b128 %0, %1, off"
                 :: "v"(lds_byte_off), "v"(gaddr)
                 : "memory");
}

__device__ __forceinline__ void wait_async_zero() {
    asm volatile("s_wait_asynccnt 0x0" ::: "memory");
}

// LDS byte offset of a generic shared pointer (low 32 bits of the flat
// address are the LDS offset on gfx1250).
__device__ __forceinline__ uint32_t lds_off(const void* p) {
    return (uint32_t)(uintptr_t)p;
}

// Workgroup: 256 threads (8 waves), macro tile 128(d) x 256(t) for one b.
// Wave (wd, wt): 64x64 output = 4x4 tiles of 16x16, K in steps of 4.
__global__ __launch_bounds__(256, 1)
void subject_layers_wmma_f32(const float* __restrict__ x,
                             const int* __restrict__ subjects,
                             const float* __restrict__ weights,
                             float* __restrict__ out)
{
    __shared__ float Wsm[2][KB * WS];   // [buf][c][d] padded
    __shared__ float Xsm[2][KB * XS];   // [buf][c][t] padded

    const int b    = blockIdx.z;
    const int dblk = blockIdx.y;        // 0..1  (128 d)
    const int tblk = blockIdx.x;        // 0..15 (256 t)

    const int tid  = threadIdx.x;
    const int wave = tid >> 5;
    const int lane = tid & 31;
    const int m    = lane & 15;         // M (A) / N (B) index
    const int kh   = lane >> 4;         // K pair select

    const int wd = wave & 1;            // 2 waves across d
    const int wt = wave >> 1;           // 4 waves across t
    const int dw = wd * 64;             // wave d offset within macro tile
    const int tw = wt * 64;             // wave t offset within macro tile

    const int s = subjects[b];
    const float* __restrict__ W = weights + (size_t)s * GC * GD;  // [C][D]
    const float* __restrict__ X = x       + (size_t)b * GC * GT;  // [C][T]
    float*       __restrict__ O = out     + (size_t)b * GD * GT;  // [D][T]

    const float* __restrict__ Wg = W + (size_t)dblk * 128;  // + c*GD + d
    const float* __restrict__ Xg = X + (size_t)tblk * 256;  // + c*GT + t

    // Issue one stage of async panel copies (all 256 threads participate).
    auto issue_stage = [&](int st, int buf) {
        const int ck = st * KB;
        // W panel: KB x 128 floats = 512 float4 -> 2 per thread.
#pragma unroll
        for (int r = 0; r < 2; ++r) {
            const int q  = tid + r * 256;
            const int c  = q >> 5;             // 32 quads per c-row
            const int dq = (q & 31) << 2;
            async_copy_b128(lds_off(&Wsm[buf][c * WS + dq]),
                            Wg + (size_t)(ck + c) * GD + dq);
        }
        // X panel: KB x 256 floats = 1024 float4 -> 4 per thread.
#pragma unroll
        for (int r = 0; r < 4; ++r) {
            const int q  = tid + r * 256;
            const int c  = q >> 6;             // 64 quads per c-row
            const int tq = (q & 63) << 2;
            async_copy_b128(lds_off(&Xsm[buf][c * XS + tq]),
                            Xg + (size_t)(ck + c) * GT + tq);
        }
    };

    v8f acc[4][4];
#pragma unroll
    for (int i = 0; i < 4; ++i)
#pragma unroll
        for (int j = 0; j < 4; ++j)
#pragma unroll
            for (int r = 0; r < 8; ++r)
                acc[i][j][r] = 0.0f;

    issue_stage(0, 0);

    for (int st = 0; st < NSTAGE; ++st) {
        // Stage st's async copies complete; all waves done reading st-1.
        wait_async_zero();
        __syncthreads();
        if (st + 1 < NSTAGE)
            issue_stage(st + 1, (st + 1) & 1);

        const float* __restrict__ Wp = Wsm[st & 1];
        const float* __restrict__ Xp = Xsm[st & 1];

#pragma unroll
        for (int kk = 0; kk < KB; kk += 4) {
            const int c2 = kk + 2 * kh;        // this lane's first K row
            v2f afrag[4];
            v2f bfrag[4];
#pragma unroll
            for (int i = 0; i < 4; ++i) {
                const int dl = dw + i * 16 + m;
                afrag[i].x = Wp[c2 * WS + dl];
                afrag[i].y = Wp[(c2 + 1) * WS + dl];
            }
#pragma unroll
            for (int j = 0; j < 4; ++j) {
                const int tl = tw + j * 16 + m;
                bfrag[j].x = Xp[c2 * XS + tl];
                bfrag[j].y = Xp[(c2 + 1) * XS + tl];
            }
#pragma unroll
            for (int i = 0; i < 4; ++i)
#pragma unroll
                for (int j = 0; j < 4; ++j)
                    acc[i][j] = __builtin_amdgcn_wmma_f32_16x16x4_f32(
                        false, afrag[i], false, bfrag[j],
                        (short)0, acc[i][j], false, false);
        }
    }

    // Store: VGPR r holds rows M=r (lanes 0-15) / r+8 (lanes 16-31); N=m.
    const int dmacro = dblk * 128;
    const int tmacro = tblk * 256;
#pragma unroll
    for (int i = 0; i < 4; ++i) {
#pragma unroll
        for (int j = 0; j < 4; ++j) {
            const int tcol = tmacro + tw + j * 16 + m;
#pragma unroll
            for (int r = 0; r < 8; ++r) {
                const int drow = dmacro + dw + i * 16 + r + 8 * kh;
                O[(size_t)drow * GT + tcol] = acc[i][j][r];
            }
        }
    }
}

extern "C" void kernel_launch(void* const* d_in, const int* in_sizes, int n_in,
                              void* d_out, int out_size, void* d_ws, size_t ws_size,
                              hipStream_t stream) {
    const float* x        = (const float*)d_in[0];
    const int*   subjects = (const int*)d_in[1];
    const float* weights  = (const float*)d_in[2];
    float*       out      = (float*)d_out;

    dim3 grid(GT / 256, GD / 128, GB);  // (16, 2, 64)
    dim3 block(256, 1, 1);
    subject_layers_wmma_f32<<<grid, block, 0, stream>>>(x, subjects, weights, out);
}